// YOLOLayer_56332791054646
// MI455X (gfx1250) — compile-verified
//
#include <hip/hip_runtime.h>
#include <stdint.h>

// YOLO head: (16, 4*65, 128, 128) f32 -> (16, 4*128*128, 65) f32
// Memory-bound transpose + sigmoid. TDM (tensor_load_to_lds) stages a
// 65(ch) x 64(spatial) tile into LDS with pad -> row stride 66 floats
// (bank-conflict-free transposed reads), then threads emit the contiguous
// channel-last output with branchless per-channel fixup and non-temporal
// coalesced stores.

#define NCLS     60
#define NANCH    4
#define BBOX     65              // 5 + NCLS
#define NG       128
#define SPATIAL  (NG * NG)       // 16384
#define TILE_S   64              // spatial positions per tile
#define LDS_ROW  66              // 64 + 2 pad DWORDs inserted by TDM pad
#define STRIDEF  8.0f            // 1024 / 128

typedef unsigned int v4u __attribute__((ext_vector_type(4)));
typedef int          v8i __attribute__((ext_vector_type(8)));
typedef int          v4i __attribute__((ext_vector_type(4)));

// bw = sigmoid(tw) * (anchor_norm*1024/stride) * 5 * stride = sigmoid * anchor_norm * 5120
__constant__ float c_aw[NANCH] = {0.05f * 5120.0f, 0.11f * 5120.0f, 0.21f * 5120.0f, 0.40f * 5120.0f};
__constant__ float c_ah[NANCH] = {0.062f * 5120.0f, 0.13f * 5120.0f, 0.26f * 5120.0f, 0.45f * 5120.0f};

__global__ __launch_bounds__(256)
void yolo_tdm_kernel(const float* __restrict__ x, float* __restrict__ out)
{
    // Only LDS allocation in this kernel -> LDS byte offset 0 (matches D#.lds_addr).
    __shared__ float tile[BBOX * LDS_ROW];   // 65 * 66 floats = 17160 B

    const int t  = blockIdx.x;               // 0 .. 16383
    const int s0 = (t & 255) * TILE_S;       // spatial tile start within 16384
    const int a  = (t >> 8) & (NANCH - 1);   // anchor
    const int b  = t >> 10;                  // batch

    // Tile start in global memory: x[(b*NANCH + a)*BBOX*SPATIAL + 0*SPATIAL + s0]
    const uint64_t gaddr =
        (uint64_t)(uintptr_t)(x + ((size_t)(b * NANCH + a) * BBOX) * SPATIAL + s0);

    if (threadIdx.x < 32) {                  // wave 0 issues the DMA once
        // ---- D# group 0 (128b): count | lds_addr | global_addr | type=2 ----
        v4u g0;
        g0.x = 1u;                                        // count=1, user descriptor
        g0.y = 0u;                                        // lds_addr = 0 (tile at LDS offset 0)
        g0.z = (uint32_t)(gaddr & 0xFFFFFFFFu);           // global_addr[31:0]
        g0.w = (uint32_t)((gaddr >> 32) & 0x01FFFFFFu)    // global_addr[56:32]
             | (2u << 30);                                // type = 2 ("image")

        // ---- D# group 1 (256b) ----
        v8i g1;
        // data_size=2 (4B), pad_enable, pad_interval=5 (64 DW), pad_amount=1 (2 DW)
        g1[0] = (int)((2u << 16) | (1u << 20) | (5u << 22) | (1u << 25));
        g1[1] = (int)(16384u << 16);   // tensor_dim0[15:0]=16384 at bits[63:48]
        g1[2] = (int)(65u << 16);      // tensor_dim0 hi=0 | tensor_dim1=65 (channels)
        g1[3] = (int)(64u << 16);      // tensor_dim1 hi=0 | tile_dim0=64 (spatial)
        g1[4] = (int)(65u);            // tile_dim1=65 | tile_dim2=0
        g1[5] = (int)(16384u);         // tensor_dim0_stride lo (elements between channel rows)
        g1[6] = 0;                     // stride0 hi | tensor_dim1_stride lo
        g1[7] = 0;                     // tensor_dim1_stride hi

        v4i gz4  = {0, 0, 0, 0};               // groups 2/3 unused (2D tile)
        v8i gz8  = {0, 0, 0, 0, 0, 0, 0, 0};   // extra group (clang-23 6-arg form)

        __builtin_amdgcn_tensor_load_to_lds(g0, g1, gz4, gz4, gz8, 0);
        __builtin_amdgcn_s_wait_tensorcnt(0);
    }
    __syncthreads();

    // The TDM wrote `tile` behind the compiler's back (LDS address is an
    // integer inside the descriptor). Escape the pointer + memory clobber so
    // the ds_loads below are not folded to undef. Executed by all waves so it
    // dominates every read.
    {
        float* p = tile;
        asm volatile("" : "+v"(p) : : "memory");
    }

    const float aw = c_aw[a];                // hoisted anchor scales
    const float ah = c_ah[a];

    // Output: contiguous 64*65 floats starting at ((b*4 + a)*16384 + s0)*65
    const size_t obase = ((size_t)(b * NANCH + a) * SPATIAL + (size_t)s0) * BBOX;

    #pragma unroll 4
    for (int j = threadIdx.x; j < TILE_S * BBOX; j += 256) {
        const int s = j / BBOX;              // spatial within tile (magic-mul)
        const int c = j - s * BBOX;          // channel 0..64
        const float v  = tile[c * LDS_ROW + s];
        const float sg = 1.0f / (1.0f + __expf(-v));   // v_exp_f32 fast path

        const int gs = s0 + s;               // global spatial index = y*128 + x

        // Branchless per-channel fixup: r = (sg + add) * mul
        //   c==0: add=grid_x, mul=8   c==1: add=grid_y, mul=8
        //   c==2: add=0, mul=aw       c==3: add=0, mul=ah
        //   c>=4: add=0, mul=1
        const float add = (c == 0) ? (float)(gs & (NG - 1))
                        : (c == 1) ? (float)(gs >> 7)
                        : 0.0f;
        const float mul = (c <  2) ? STRIDEF
                        : (c == 2) ? aw
                        : (c == 3) ? ah
                        : 1.0f;
        const float r = (sg + add) * mul;

        __builtin_nontemporal_store(r, &out[obase + (size_t)j]);
    }
}

extern "C" void kernel_launch(void* const* d_in, const int* in_sizes, int n_in,
                              void* d_out, int out_size, void* d_ws, size_t ws_size,
                              hipStream_t stream) {
    (void)in_sizes; (void)n_in; (void)out_size; (void)d_ws; (void)ws_size;
    const float* x  = (const float*)d_in[0];
    float* out      = (float*)d_out;

    const int n_tiles = 16 * NANCH * (SPATIAL / TILE_S);   // 16384 blocks
    yolo_tdm_kernel<<<dim3(n_tiles), dim3(256), 0, stream>>>(x, out);
}